// LatentSDE_26431228740114
// MI455X (gfx1250) — compile-verified
//
#include <hip/hip_runtime.h>
#include <math.h>

// LatentSDE forward for MI455X (gfx1250, wave32).
// k_sde: one wave = 32 distinct batch elements (batch = lane). The 8x8 hidden
// layers of the f- and h-MLPs are fused into a stacked block-diagonal
// V_WMMA_F32_16X16X4_F32 chain: 2 batch groups x 4 K-chunks = 8 WMMAs/substep.
// Cross-layout transposes go through a bank-conflict-padded LDS staging buffer.
// Transcendentals use raw v_exp_f32/v_log_f32/v_rcp_f32 (inputs never denormal
// where it matters), avoiding libm range-fixup code.

#define B_N    65536
#define T_N    24
#define D_N    2
#define L_N    2
#define H_N    8
#define NSUB_N 10
#define HALF_LOG_2PI 0.9189385332046727f
#define LOG2E_F 1.4426950408889634f
#define LN2_F   0.6931471805599453f

typedef __attribute__((ext_vector_type(2))) float v2f;
typedef __attribute__((ext_vector_type(8))) float v8f;

__device__ __forceinline__ float lane_readf(float v, int byteIdx) {
  return __int_as_float(__builtin_amdgcn_ds_bpermute(byteIdx, __float_as_int(v)));
}
__device__ __forceinline__ float fast_exp(float x) {            // e^x
  return __builtin_amdgcn_exp2f(x * LOG2E_F);
}
__device__ __forceinline__ float softplusf(float x) {           // ln(1+e^x)
  const float e = __builtin_amdgcn_exp2f(fabsf(x) * -LOG2E_F);
  return fmaf(LN2_F, __builtin_amdgcn_logf(1.0f + e), fmaxf(x, 0.0f));
}

// Pointwise encoder MLP: 2 -> 8 -> 8 -> 1 (relu).
__device__ __forceinline__ float enc_eval(
    const float* __restrict__ w1, const float* __restrict__ b1,
    const float* __restrict__ w2, const float* __restrict__ b2,
    const float* __restrict__ w3, const float* __restrict__ b3,
    float x0, float x1) {
  float h1[H_N];
#pragma unroll
  for (int j = 0; j < H_N; j++)
    h1[j] = fmaxf(w1[j * 2] * x0 + w1[j * 2 + 1] * x1 + b1[j], 0.0f);
  float h2[H_N];
#pragma unroll
  for (int j = 0; j < H_N; j++) {
    float s = b2[j];
#pragma unroll
    for (int i = 0; i < H_N; i++) s += w2[j * 8 + i] * h1[i];
    h2[j] = fmaxf(s, 0.0f);
  }
  float o = b3[0];
#pragma unroll
  for (int j = 0; j < H_N; j++) o += w3[j] * h2[j];
  return o;
}

// ---------------------------------------------------------------------------
__global__ void k_init(float* acc) {
  if (threadIdx.x == 0) { acc[0] = 0.0f; acc[1] = 0.0f; acc[2] = 0.0f; }
}

// ---------------------------------------------------------------------------
// Encoder at t=0 -> q(z0), z0 sample, KL term. One thread per batch element.
__global__ void k_encoder(
    const float* __restrict__ ew1, const float* __restrict__ eb1,
    const float* __restrict__ ew2, const float* __restrict__ eb2,
    const float* __restrict__ ew3, const float* __restrict__ eb3,
    const float* __restrict__ qw,  const float* __restrict__ qb,
    const float* __restrict__ pm,  const float* __restrict__ pls,
    const float* __restrict__ xs,  const float* __restrict__ eps,
    float* __restrict__ zs, float* __restrict__ acc) {
  const int b = blockIdx.x * blockDim.x + threadIdx.x;
  const v2f xv = reinterpret_cast<const v2f*>(xs)[b];  // xs[0,b,:]
  const float c0 = enc_eval(ew1, eb1, ew2, eb2, ew3, eb3, xv.x, xv.y);
  const float qm0 = qw[0] * c0 + qb[0];
  const float qm1 = qw[1] * c0 + qb[1];
  const float ql0 = qw[2] * c0 + qb[2];
  const float ql1 = qw[3] * c0 + qb[3];
  const v2f ev = reinterpret_cast<const v2f*>(eps)[b];
  v2f z; z.x = qm0 + fast_exp(ql0) * ev.x; z.y = qm1 + fast_exp(ql1) * ev.y;
  reinterpret_cast<v2f*>(zs)[b] = z;  // zs[0,b,:]

  float kl = 0.0f;
  {
    const float i2p0 = 0.5f * __builtin_amdgcn_rcpf(fast_exp(2.0f * pls[0]));
    const float i2p1 = 0.5f * __builtin_amdgcn_rcpf(fast_exp(2.0f * pls[1]));
    const float d0 = qm0 - pm[0], d1 = qm1 - pm[1];
    kl += pls[0] - ql0 + (fast_exp(2.0f * ql0) + d0 * d0) * i2p0 - 0.5f;
    kl += pls[1] - ql1 + (fast_exp(2.0f * ql1) + d1 * d1) * i2p1 - 0.5f;
  }
  __shared__ float red[256];
  red[threadIdx.x] = kl;
  __syncthreads();
#pragma unroll
  for (int s = 128; s > 0; s >>= 1) {
    if ((int)threadIdx.x < s) red[threadIdx.x] += red[threadIdx.x + s];
    __syncthreads();
  }
  if (threadIdx.x == 0) atomicAdd(&acc[1], red[0]);
}

// ---------------------------------------------------------------------------
// SDE integration: 32 distinct batch elements per wave (batch = lane).
__global__ void __launch_bounds__(256) k_sde(
    const float* __restrict__ fw1, const float* __restrict__ fb1,
    const float* __restrict__ fw2, const float* __restrict__ fb2,
    const float* __restrict__ fw3, const float* __restrict__ fb3,
    const float* __restrict__ hw1, const float* __restrict__ hb1,
    const float* __restrict__ hw2, const float* __restrict__ hb2,
    const float* __restrict__ hw3, const float* __restrict__ hb3,
    const float* __restrict__ gw1, const float* __restrict__ gb1,
    const float* __restrict__ gw2, const float* __restrict__ gb2v,
    const float* __restrict__ ew1, const float* __restrict__ eb1,
    const float* __restrict__ ew2, const float* __restrict__ eb2,
    const float* __restrict__ ew3, const float* __restrict__ eb3,
    const float* __restrict__ ts,  const float* __restrict__ xs,
    const float* __restrict__ dW,  float* __restrict__ zs,
    float* __restrict__ acc) {
  const int b    = blockIdx.x * blockDim.x + threadIdx.x;  // batch = lane
  const int lane = threadIdx.x & 31;
  const int n    = lane & 15;
  const int hi   = lane >> 4;          // 0 = low half (f rows), 1 = high (h)
  const bool low = (hi == 0);
  const int w    = threadIdx.x >> 5;   // wave in block

  // LDS staging. hid: batch-major, 20-dword padded stride so the 16 gather
  // source lanes hit 16 distinct 4-bank windows (conflict-free b64 gathers).
  __shared__ float s_hid[8][32][20];
  __shared__ float s_o[8][32][4];
  float* const hidb  = &s_hid[w][0][0];
  float* const myhid = hidb + lane * 20;
  const float* const bl0 = hidb + n * 20 + hi * 2;         // batch group 0
  const float* const bl1 = hidb + (16 + n) * 20 + hi * 2;  // batch group 1
  float* const ob  = &s_o[w][0][0];
  float* const myo = ob + lane * 4;
  const float* const fvp = ob + n * 4 + hi * 2;            // f from src lane n
  const float* const hvp = ob + (16 + n) * 4 + hi * 2;     // h from lane n+16

  // A-matrix lane mapping for 16x16x4 f32 WMMA:
  //   VGPR0: lanes 0-15 -> (M=lane, K=kb), lanes 16-31 -> (M=lane-16, K=kb)
  const int m  = n;
  const int kb = hi * 2;

  // Stacked block-diagonal layer-2 weights: rows 0-7 = f_w2, rows 8-15 = h_w2,
  // K 0-7 = f-hidden, K 8-15 = h-hidden; 4 chunks of K=4.
  v2f aF0, aF1, aH0, aH1;
  aF0.x = (m < 8) ? fw2[m * 8 + kb]         : 0.0f;
  aF0.y = (m < 8) ? fw2[m * 8 + kb + 1]     : 0.0f;
  aF1.x = (m < 8) ? fw2[m * 8 + 4 + kb]     : 0.0f;
  aF1.y = (m < 8) ? fw2[m * 8 + 4 + kb + 1] : 0.0f;
  aH0.x = (m >= 8) ? hw2[(m - 8) * 8 + kb]         : 0.0f;
  aH0.y = (m >= 8) ? hw2[(m - 8) * 8 + kb + 1]     : 0.0f;
  aH1.x = (m >= 8) ? hw2[(m - 8) * 8 + 4 + kb]     : 0.0f;
  aH1.y = (m >= 8) ? hw2[(m - 8) * 8 + 4 + kb + 1] : 0.0f;

  v8f ci;  // layer-2 bias in C layout (rows r / r+8 split across half-waves)
#pragma unroll
  for (int r = 0; r < 8; r++) ci[r] = low ? fb2[r] : hb2[r];

  // Layer-3 weights in C layout: f_w3 for low half, h_w3 for high half.
  float w3s0[8], w3s1[8];
#pragma unroll
  for (int k = 0; k < 8; k++) {
    w3s0[k] = low ? fw3[k]     : hw3[k];
    w3s1[k] = low ? fw3[8 + k] : hw3[8 + k];
  }
  const float b3s0 = low ? fb3[0] : hb3[0];
  const float b3s1 = low ? fb3[1] : hb3[1];
  const float spmul = low ? 0.0f : LN2_F;  // softplus log-term only on h rows

  v2f zv = reinterpret_cast<const v2f*>(zs)[b];  // z0 from encoder kernel
  float z0 = zv.x, z1 = zv.y;
  float path = 0.0f;

#pragma unroll 1
  for (int i = 0; i < T_N - 1; i++) {
    // ctx[i+1] (searchsorted resolves to i+1 inside interval i)
    const v2f xv = reinterpret_cast<const v2f*>(xs)[(size_t)(i + 1) * B_N + b];
    const float c = enc_eval(ew1, eb1, ew2, eb2, ew3, eb3, xv.x, xv.y);
    const float dt = (ts[i + 1] - ts[i]) * (1.0f / NSUB_N);
    const float sqdt = sqrtf(dt);
    float lr = 0.0f;

#pragma unroll 1
    for (int ks = 0; ks < NSUB_N; ks++) {
      // Layer 1 on VALU, batch-per-lane: hid[0..7] = f relu, hid[8..15] = h sp.
      float hid[16];
#pragma unroll
      for (int j = 0; j < 8; j++) {
        hid[j]     = fmaxf(fw1[j * 3] * z0 + fw1[j * 3 + 1] * z1 +
                           fw1[j * 3 + 2] * c + fb1[j], 0.0f);
        hid[8 + j] = softplusf(hw1[j * 2] * z0 + hw1[j * 2 + 1] * z1 + hb1[j]);
      }
      // Stage hid to LDS (4x ds_store_b128, k-contiguous per batch).
      reinterpret_cast<float4*>(myhid)[0] = make_float4(hid[0], hid[1], hid[2], hid[3]);
      reinterpret_cast<float4*>(myhid)[1] = make_float4(hid[4], hid[5], hid[6], hid[7]);
      reinterpret_cast<float4*>(myhid)[2] = make_float4(hid[8], hid[9], hid[10], hid[11]);
      reinterpret_cast<float4*>(myhid)[3] = make_float4(hid[12], hid[13], hid[14], hid[15]);
      asm volatile("" ::: "memory");  // same-wave LDS ops are in-order in HW

      // Layer 2: two batch groups x 4 chained f32 WMMAs; each B operand is one
      // conflict-free ds_load_b64 gather from the staged hid.
      v8f am0 = ci, am1 = ci;
      v2f bb;
      bb = *reinterpret_cast<const v2f*>(bl0 + 0);
      am0 = __builtin_amdgcn_wmma_f32_16x16x4_f32(false, aF0, false, bb, (short)0, am0, false, false);
      bb = *reinterpret_cast<const v2f*>(bl1 + 0);
      am1 = __builtin_amdgcn_wmma_f32_16x16x4_f32(false, aF0, false, bb, (short)0, am1, false, false);
      bb = *reinterpret_cast<const v2f*>(bl0 + 4);
      am0 = __builtin_amdgcn_wmma_f32_16x16x4_f32(false, aF1, false, bb, (short)0, am0, false, false);
      bb = *reinterpret_cast<const v2f*>(bl1 + 4);
      am1 = __builtin_amdgcn_wmma_f32_16x16x4_f32(false, aF1, false, bb, (short)0, am1, false, false);
      bb = *reinterpret_cast<const v2f*>(bl0 + 8);
      am0 = __builtin_amdgcn_wmma_f32_16x16x4_f32(false, aH0, false, bb, (short)0, am0, false, false);
      bb = *reinterpret_cast<const v2f*>(bl1 + 8);
      am1 = __builtin_amdgcn_wmma_f32_16x16x4_f32(false, aH0, false, bb, (short)0, am1, false, false);
      bb = *reinterpret_cast<const v2f*>(bl0 + 12);
      am0 = __builtin_amdgcn_wmma_f32_16x16x4_f32(false, aH1, false, bb, (short)0, am0, false, false);
      bb = *reinterpret_cast<const v2f*>(bl1 + 12);
      am1 = __builtin_amdgcn_wmma_f32_16x16x4_f32(false, aH1, false, bb, (short)0, am1, false, false);

      // Layer-2 activation + layer-3 in C layout per group:
      // relu rows on lanes 0-15 (f), softplus rows on lanes 16-31 (h).
      // softplus(v) = max(v,0) + ln2*log2(1 + 2^(-|v|*log2e)); the log term is
      // masked off on f rows via spmul (0 or ln2).
      float o0g0 = b3s0, o1g0 = b3s1, o0g1 = b3s0, o1g1 = b3s1;
#pragma unroll
      for (int r = 0; r < 8; r++) {
        const float v0 = am0[r];
        const float e0 = __builtin_amdgcn_exp2f(fabsf(v0) * -LOG2E_F);
        const float a0 = fmaf(spmul, __builtin_amdgcn_logf(1.0f + e0),
                              fmaxf(v0, 0.0f));
        const float v1 = am1[r];
        const float e1 = __builtin_amdgcn_exp2f(fabsf(v1) * -LOG2E_F);
        const float a1 = fmaf(spmul, __builtin_amdgcn_logf(1.0f + e1),
                              fmaxf(v1, 0.0f));
        o0g0 += w3s0[r] * a0; o1g0 += w3s1[r] * a0;
        o0g1 += w3s0[r] * a1; o1g1 += w3s1[r] * a1;
      }
      // Redistribute f/h back to batch-per-lane via LDS (store b128, 2x b64).
      reinterpret_cast<float4*>(myo)[0] = make_float4(o0g0, o1g0, o0g1, o1g1);
      asm volatile("" ::: "memory");
      const v2f fv = *reinterpret_cast<const v2f*>(fvp);  // (f0, f1) this batch
      const v2f hv = *reinterpret_cast<const v2f*>(hvp);  // (h0, h1) this batch

      // Diffusion g (per latent dim, batch-per-lane).
      float s0 = gb2v[0], s1 = gb2v[1];
#pragma unroll
      for (int j = 0; j < 8; j++) {
        s0 += gw2[j]     * softplusf(z0 * gw1[j]     + gb1[j]);
        s1 += gw2[8 + j] * softplusf(z1 * gw1[8 + j] + gb1[8 + j]);
      }
      // g = sigmoid(s) = rcp(1 + 2^(-s*log2e)); 1/g = (1 + e) directly, so the
      // drift-ratio u = (f-h)/g needs no division at all.
      const float t0 = 1.0f + __builtin_amdgcn_exp2f(s0 * -LOG2E_F);
      const float t1 = 1.0f + __builtin_amdgcn_exp2f(s1 * -LOG2E_F);
      const float g0 = __builtin_amdgcn_rcpf(t0);
      const float g1 = __builtin_amdgcn_rcpf(t1);

      const float u0 = (fv.x - hv.x) * t0;
      const float u1 = (fv.y - hv.y) * t1;
      lr += 0.5f * dt * (u0 * u0 + u1 * u1);

      const v2f dw = __builtin_nontemporal_load(
          reinterpret_cast<const v2f*>(dW) + ((size_t)(i * NSUB_N + ks) * B_N + b));
      z0 += fv.x * dt + g0 * sqdt * dw.x;
      z1 += fv.y * dt + g1 * sqdt * dw.y;
    }
    v2f zo; zo.x = z0; zo.y = z1;
    reinterpret_cast<v2f*>(zs)[(size_t)(i + 1) * B_N + b] = zo;
    path += lr;
  }
  // Full-wave reduction (32 distinct batch elements).
  float s = path;
#pragma unroll
  for (int msk = 1; msk < 32; msk <<= 1) s += lane_readf(s, (lane ^ msk) << 2);
  if (lane == 0) atomicAdd(&acc[2], s);
}

// ---------------------------------------------------------------------------
// Projection + log-likelihood. One thread per (t, b).
__global__ void k_proj(
    const float* __restrict__ pw1, const float* __restrict__ pb1,
    const float* __restrict__ pw2, const float* __restrict__ pb2,
    const float* __restrict__ xs,  const float* __restrict__ ns_p,
    const float* __restrict__ zs,  float* __restrict__ xs_out,
    float* __restrict__ acc) {
  const int idx = blockIdx.x * blockDim.x + threadIdx.x;  // t*B + b
  const v2f z = reinterpret_cast<const v2f*>(zs)[idx];
  float hp[8];
#pragma unroll
  for (int j = 0; j < 8; j++)
    hp[j] = fmaxf(pw1[j * 2] * z.x + pw1[j * 2 + 1] * z.y + pb1[j], 0.0f);
  float o0 = pb2[0], o1 = pb2[1];
#pragma unroll
  for (int j = 0; j < 8; j++) { o0 += pw2[j] * hp[j]; o1 += pw2[8 + j] * hp[j]; }
  v2f ov; ov.x = o0; ov.y = o1;
  reinterpret_cast<v2f*>(xs_out)[idx] = ov;

  const float ns = ns_p[0];
  const float inv = __builtin_amdgcn_rcpf(ns);
  const float logns = LN2_F * __builtin_amdgcn_logf(ns);
  const v2f xv = reinterpret_cast<const v2f*>(xs)[idx];
  const float d0 = (xv.x - o0) * inv, d1 = (xv.y - o1) * inv;
  const float lp = -0.5f * (d0 * d0 + d1 * d1) - 2.0f * (logns + HALF_LOG_2PI);

  __shared__ float red[256];
  red[threadIdx.x] = lp;
  __syncthreads();
#pragma unroll
  for (int s = 128; s > 0; s >>= 1) {
    if ((int)threadIdx.x < s) red[threadIdx.x] += red[threadIdx.x + s];
    __syncthreads();
  }
  if (threadIdx.x == 0) atomicAdd(&acc[0], red[0]);
}

// ---------------------------------------------------------------------------
__global__ void k_final(float* __restrict__ out, const float* __restrict__ acc) {
  if (threadIdx.x == 0) {
    out[0] = acc[0] * (1.0f / B_N);                 // log_pxs
    out[1] = (acc[1] + acc[2]) * (1.0f / B_N);      // logqp0 + logqp_path
  }
}

// ---------------------------------------------------------------------------
extern "C" void kernel_launch(void* const* d_in, const int* in_sizes, int n_in,
                              void* d_out, int out_size, void* d_ws, size_t ws_size,
                              hipStream_t stream) {
  (void)in_sizes; (void)out_size; (void)ws_size;
  // Param leaves flattened in setup_inputs() dict insertion order, then
  // xs, ts, noise_std, eps_z0, dW.
  const float* P[35];
  for (int i = 0; i < 35 && i < n_in; i++) P[i] = (const float*)d_in[i];

  float* out    = (float*)d_out;
  float* xs_out = out + 2;                               // _xs : (T,B,D)
  float* zs_out = out + 2 + (size_t)T_N * B_N * D_N;     // zs  : (T,B,L)
  float* acc    = (float*)d_ws;                          // 3 accumulators

  k_init<<<1, 1, 0, stream>>>(acc);

  k_encoder<<<B_N / 256, 256, 0, stream>>>(
      P[0], P[1], P[2], P[3], P[4], P[5],   // enc
      P[6], P[7],                           // qz0
      P[28], P[29],                         // pz0 mean / logstd
      P[30], P[33],                         // xs, eps_z0
      zs_out, acc);

  k_sde<<<B_N / 256, 256, 0, stream>>>(
      P[8],  P[9],  P[10], P[11], P[12], P[13],   // f
      P[14], P[15], P[16], P[17], P[18], P[19],   // h
      P[20], P[21], P[22], P[23],                 // g
      P[0],  P[1],  P[2],  P[3],  P[4],  P[5],    // enc (ctx inline)
      P[31], P[30], P[34],                        // ts, xs, dW
      zs_out, acc);

  k_proj<<<(T_N * B_N) / 256, 256, 0, stream>>>(
      P[24], P[25], P[26], P[27],                 // proj
      P[30], P[32],                               // xs, noise_std
      zs_out, xs_out, acc);

  k_final<<<1, 1, 0, stream>>>(out, acc);
}